// PositionIntervalLoss_11209864643165
// MI455X (gfx1250) — compile-verified
//
#include <hip/hip_runtime.h>
#include <stdint.h>

// ---------- types for WMMA / TDM builtins ----------
typedef float v2f  __attribute__((ext_vector_type(2)));
typedef float v8f  __attribute__((ext_vector_type(8)));
typedef unsigned int u32v4 __attribute__((ext_vector_type(4)));
typedef int   i32v4 __attribute__((ext_vector_type(4)));
typedef int   i32v8 __attribute__((ext_vector_type(8)));

namespace {
constexpr int NF    = 300;  // row length
constexpr int ROWS  = 16;   // rows per block (one WMMA M-tile)
constexpr int PAD   = 4;    // zeroed left pad per input row (conv halo)
constexpr int LSTR  = 313;  // LDS row stride (floats); gcd(313,64)=1 -> no bank conflicts
                            // layout per input row: [0..3]=0, [4..303]=data, [304..312]=0
constexpr int PCAP  = 152;  // >= PMAX = F/2+1 = 151
}

// Gaussian kernel, sigma=2.0, length round(3*sigma+1)=7, normalized
// (precomputed to match reference _gauss_kernel in f32)
__device__ __constant__ float GK[7] = {
    0.07015933f, 0.13107467f, 0.19071269f, 0.21610594f,
    0.19071269f, 0.13107467f, 0.07015933f};

__device__ __forceinline__ float gk_tap(int t) {
  return (t >= 0 && t < 7) ? GK[t] : 0.0f;
}

__global__ __launch_bounds__(32)
void PositionIntervalLoss_kernel(const float* __restrict__ fr,
                                 float* __restrict__ out, int Btot) {
  __shared__ float          s_in  [ROWS * LSTR];   // padded staged input rows
  __shared__ float          s_blur[ROWS * LSTR];   // blurred rows (+slack for tail tile)
  __shared__ unsigned short s_pk  [ROWS * PCAP];   // peak positions per row

  const int lane = threadIdx.x;          // 0..31 (wave32)
  const int h    = (lane >> 4) & 1;      // lane half
  const int m    = lane & 15;            // M (row) for A, N (col) for B/D
  const int row0 = blockIdx.x * ROWS;
  const int nrows = min(ROWS, Btot - row0);

  // Zero the conv halo pads (13 floats per row) -- turns boundary handling
  // into data instead of control flow inside the WMMA loop.
  for (int i = lane; i < ROWS * 13; i += 32) {
    const int r = i / 13, k = i % 13;
    s_in[r * LSTR + ((k < PAD) ? k : (NF + k))] = 0.0f;   // [0..3] and [304..312]
  }

  // ================= Phase 1: TDM global -> LDS (stride-313 padded rows) =====
#if __has_builtin(__builtin_amdgcn_tensor_load_to_lds)
  {
    uint64_t gaddr = (uint64_t)(uintptr_t)(fr + (size_t)row0 * NF);
    // flat LDS pointer truncates to byte offset within wave's LDS (aperture rule)
    uint32_t lds_addr = (uint32_t)(uintptr_t)(&s_in[PAD]);

    u32v4 g0;
    g0[0] = 1u;                                        // count=1, no gather
    g0[1] = lds_addr;                                  // lds_addr (bytes)
    g0[2] = (uint32_t)gaddr;                           // global_addr[31:0]
    g0[3] = (uint32_t)((gaddr >> 32) & 0x01FFFFFFu)    // global_addr[56:32]
            | (2u << 30);                              // type=2 ("image")

    i32v8 g1;
    g1[0] = (2 << 16) | (1 << 19);   // data_size=4B, iterate_enable=1, mask=0
    g1[1] = (NF & 0xFFFF) << 16;     // tensor_dim0 = 300 (lo16)
    g1[2] = (1 << 16);               // tensor_dim0 hi=0 | tensor_dim1 = 1 (lo16)
    g1[3] = (NF & 0xFFFF) << 16;     // tensor_dim1 hi=0 | tile_dim0 = 300
    g1[4] = 1;                       // tile_dim1 = 1 | tile_dim2 = 0
    g1[5] = NF;                      // tensor_dim0_stride = 300 (lo32)
    g1[6] = 0;                       // stride hi | tensor_dim1_stride lo
    g1[7] = 0;

    i32v4 g2;                        // iterate mode: one row per iteration
    g2[0] = 0;                       // tensor_dim2 (unused)
    g2[1] = LSTR * 4;                // lds_addr_increment  (bytes)
    g2[2] = NF * 4;                  // global_addr_increment (bytes, lo32)
    g2[3] = ((nrows - 1) & 0xFFFF) << 16;  // inc hi=0 | iterate_count = nrows-1

    i32v4 g3 = {0, 0, 0, 0};

#if defined(__clang_major__) && (__clang_major__ >= 23)
    i32v8 gpad = {0, 0, 0, 0, 0, 0, 0, 0};
    __builtin_amdgcn_tensor_load_to_lds(g0, g1, g2, g3, gpad, 0);
#else
    __builtin_amdgcn_tensor_load_to_lds(g0, g1, g2, g3, 0);
#endif
  }
#if __has_builtin(__builtin_amdgcn_s_wait_tensorcnt)
  __builtin_amdgcn_s_wait_tensorcnt(0);
#endif
#else
  // Fallback: cooperative load
  for (int i = lane; i < nrows * NF; i += 32)
    s_in[(i / NF) * LSTR + PAD + (i % NF)] = fr[(size_t)row0 * NF + i];
  __syncthreads();
#endif

  // ================= Phase 2: Gaussian blur via V_WMMA_F32_16X16X4_F32 =======
  // D(16 rows x 16 positions) = A(16 x 24 input slab) * B(24 x 16 Toeplitz),
  // 6 chained K=4 WMMAs. Exact f32 conv; zero 'SAME' padding lives in the LDS
  // pads, so the whole tile loop is branch-free (EXEC stays all-ones).
  //
  // Per-lane B operand for chunk c: b.x = B_c[2h+0][n], b.y = B_c[2h+1][n],
  // where B_c[k][n] = GK[4c + k - n - 1]  (tile-independent -> hoisted).
  v2f Bm[6];
#pragma unroll
  for (int c = 0; c < 6; ++c) {
    Bm[c].x = gk_tap(4 * c + 2 * h     - m - 1);
    Bm[c].y = gk_tap(4 * c + 2 * h + 1 - m - 1);
  }

  // A element index: s_in[m*LSTR + PAD + (j0 - 4 + 4c + 2h)] = arow[j0 + 4c]
  const float* arow = &s_in[m * LSTR + 2 * h];          // PAD - 4 == 0
  float* brow = &s_blur[(8 * h) * LSTR + m];            // D store base

#pragma unroll 1
  for (int j0 = 0; j0 < NF; j0 += 16) {
    // Batch all 12 A-loads first so the scheduler can overlap them with WMMA.
    float av[12];
    const float* ap = arow + j0;
#pragma unroll
    for (int c = 0; c < 6; ++c) {
      av[2 * c + 0] = ap[4 * c + 0];
      av[2 * c + 1] = ap[4 * c + 1];
    }
    v8f acc = {0.f, 0.f, 0.f, 0.f, 0.f, 0.f, 0.f, 0.f};
#pragma unroll
    for (int c = 0; c < 6; ++c) {
      v2f a; a.x = av[2 * c]; a.y = av[2 * c + 1];
      acc = __builtin_amdgcn_wmma_f32_16x16x4_f32(
          /*neg_a=*/false, a, /*neg_b=*/false, Bm[c],
          /*c_mod=*/(short)0, acc, /*reuse_a=*/false, /*reuse_b=*/false);
    }
    // D layout: VGPR r -> M = r + 8h, lane%16 -> N = m. Unconditional store:
    // j = j0+m <= 303 lands in the 313-float row slack; never read past 299.
#pragma unroll
    for (int r = 0; r < 8; ++r)
      brow[r * LSTR + j0] = acc[r];
  }

  // ================= Phase 3: per-row peaks + segment softargmax =============
  // Single wave: LDS ops are in-order per wave, no barrier needed.
  if (lane < ROWS && lane < nrows) {
    const int r = lane;
    const float* bl = &s_blur[r * LSTR];
    unsigned short* pk = &s_pk[r * PCAP];

    // Pass A: strict interior maxima + boundary peaks (order = sorted)
    int P = 0;
    float fp = bl[0], fc = bl[1];
    if (fp > fc) pk[P++] = 0;
#pragma unroll 1
    for (int j = 1; j < NF - 1; ++j) {
      const float fn = bl[j + 1];
      if (fc > fp && fc > fn) pk[P++] = (unsigned short)j;
      fp = fc; fc = fn;
    }
    if (fc > fp) pk[P++] = (unsigned short)(NF - 1);

    // Pass B: stream segments [segStart, segEnd) over elements 0..NF-2,
    // online stable softmax-weighted frequency per segment.
    const int nseg = (P > 0) ? P : 1;
    float sum_in = 0.0f;
    bool  any_in = false;
    float t1 = -INFINITY, t2 = -INFINITY;   // top-2 of -(sm^2) among valid
    int   segStart = 0;
#pragma unroll 1
    for (int s = 0; s < nseg; ++s) {
      const int segEnd = (s < P - 1) ? (((int)pk[s] + (int)pk[s + 1]) >> 1)
                                     : (NF - 1);
      float mx = -INFINITY, den = 0.0f, num = 0.0f;
#pragma unroll 1
      for (int j = segStart; j < segEnd; ++j) {
        const float v  = bl[j];
        const float fq = (float)(j - 99) * 0.02f - 1.0f;  // normalized freq
        if (v > mx) {
          const float sc = __expf(mx - v);   // exp(-inf)=0 on first element
          den = den * sc + 1.0f;
          num = num * sc + fq;
          mx  = v;
        } else {
          const float w = __expf(v - mx);
          den += w;
          num += w * fq;
        }
      }
      const float sm = (den > 0.0f) ? (num / den) : 0.0f;
      const float nq = -sm * sm;
      if (sm > -1.0f && sm < 1.0f) { sum_in += nq; any_in = true; }
      if (nq > t1)      { t2 = t1; t1 = nq; }
      else if (nq > t2) { t2 = nq; }
      segStart = segEnd;
    }
    const float top2sum = (t1 > -INFINITY ? t1 : 0.0f) +
                          (t2 > -INFINITY ? t2 : 0.0f);
    out[row0 + r] = any_in ? sum_in : top2sum;   // SCALE = 1.0
  }
}

extern "C" void kernel_launch(void* const* d_in, const int* in_sizes, int n_in,
                              void* d_out, int out_size, void* d_ws, size_t ws_size,
                              hipStream_t stream) {
  (void)n_in; (void)d_ws; (void)ws_size; (void)out_size;
  const float* fr = (const float*)d_in[0];
  float* out = (float*)d_out;
  const int Btot = in_sizes[0] / NF;                 // 65536
  const int grid = (Btot + ROWS - 1) / ROWS;         // 4096 blocks, 1 wave each
  PositionIntervalLoss_kernel<<<grid, 32, 0, stream>>>(fr, out, Btot);
}